// QLSTM_65481071402084
// MI455X (gfx1250) — compile-verified
//
#include <hip/hip_runtime.h>
#include <math.h>

// Problem constants (match reference setup_inputs): T=64, B=256, D=512, NQ=6
#define TT 64
#define BB 256
#define DD 512
#define NQ 6
#define NG 24           // 4 gates * 6 wires
#define NPAD 32         // padded N for two 16-wide WMMA tiles
#define MROWS (TT*BB)   // 16384

typedef float v2f __attribute__((ext_vector_type(2)));
typedef float v8f __attribute__((ext_vector_type(8)));

// ---------------------------------------------------------------------------
// Kernel 1: pack the x-part of the 4 gate weight matrices (rows 0..511 of each
// (518,6) matrix) into a transposed, padded buffer Bt[g][k], g in [0,32), k in
// [0,512). Rows 24..31 are zero padding so the second WMMA N-tile is benign.
// ---------------------------------------------------------------------------
__global__ void qlstm_pack_w(const float* __restrict__ Wf,
                             const float* __restrict__ Wi,
                             const float* __restrict__ Wu,
                             const float* __restrict__ Wo,
                             float* __restrict__ Bt) {
  int idx = blockIdx.x * blockDim.x + threadIdx.x;   // 0..16383
  int g = idx >> 9;          // padded gate-wire index 0..31
  int k = idx & (DD - 1);    // 0..511
  float v = 0.0f;
  if (g < NG) {
    int gate = g / NQ, wire = g % NQ;
    const float* W = (gate == 0) ? Wf : (gate == 1) ? Wi : (gate == 2) ? Wu : Wo;
    v = W[k * NQ + wire];    // W is (518,6) row-major; x-part rows 0..511
  }
  Bt[g * DD + k] = v;
}

// ---------------------------------------------------------------------------
// Kernel 2: P[m][g] = sum_k X[m][k] * Bt[g][k]  via V_WMMA_F32_16X16X4_F32.
// One wave per 16-row M tile, two 16-col N tiles (N padded to 32).
// A frag (16x4): lanes 0-15 hold rows M0..15 cols {k,k+1} in v0,v1;
// lanes 16-31 hold cols {k+2,k+3}.  B frag mirrors with N across lanes.
// ---------------------------------------------------------------------------
__global__ void __launch_bounds__(128)
qlstm_gemm(const float* __restrict__ X,    // (16384, 512)
           const float* __restrict__ Bt,   // (32, 512)
           float* __restrict__ P) {        // (16384, 32)
  const int wave  = (blockIdx.x * blockDim.x + threadIdx.x) >> 5;  // 0..1023
  const int lane  = threadIdx.x & 31;
  const int half  = lane >> 4;        // 0: K0/K1 slice, 1: K2/K3 slice
  const int nlane = lane & 15;
  const int row   = wave * 16 + nlane;           // A row for this lane
  const int klo   = half * 2;

  const float* arow  = X  + (size_t)row * DD + klo;
  const float* b0row = Bt + (size_t)nlane        * DD + klo;  // N tile 0
  const float* b1row = Bt + (size_t)(16 + nlane) * DD + klo;  // N tile 1

  v8f acc0 = {}; v8f acc1 = {};
  for (int k = 0; k < DD; k += 4) {
    v2f a  = *(const v2f*)(arow  + k);
    v2f b0 = *(const v2f*)(b0row + k);
    v2f b1 = *(const v2f*)(b1row + k);
    acc0 = __builtin_amdgcn_wmma_f32_16x16x4_f32(false, a, false, b0,
                                                 (short)0, acc0, false, false);
    acc1 = __builtin_amdgcn_wmma_f32_16x16x4_f32(false, a, false, b1,
                                                 (short)0, acc1, false, false);
  }

  // C/D layout: VGPR j -> row (j + 8*half), col = nlane (+16 for tile 1)
  const int mbase = wave * 16 + half * 8;
#pragma unroll
  for (int j = 0; j < 8; ++j) {
    float* prow = P + (size_t)(mbase + j) * NPAD;
    prow[nlane]      = acc0[j];
    prow[16 + nlane] = acc1[j];
  }
}

// ---------------------------------------------------------------------------
// Kernel 3: sequential LSTM scan.  One thread per batch element; h,c live in
// registers across the T=64 steps.  The qlayer is evaluated in closed form:
//   z_w = cos(pre_w + theta_w)
//   out = [z1z2z3z4z5, z0z1, z0z1z2, z0z1z2z3, z0z1z2z3z4, z0z1z2z3z4z5]
// ---------------------------------------------------------------------------
__device__ __forceinline__ float sigmoidf_(float x) {
  return 1.0f / (1.0f + expf(-x));
}

__global__ void __launch_bounds__(BB)
qlstm_scan(const float* __restrict__ P,     // (16384, 32) x-part pre-activations
           const float* __restrict__ Wf, const float* __restrict__ bf,
           const float* __restrict__ Wi, const float* __restrict__ bi,
           const float* __restrict__ Wu, const float* __restrict__ bu,
           const float* __restrict__ Wo, const float* __restrict__ bo,
           const float* __restrict__ thf, const float* __restrict__ thi,
           const float* __restrict__ thu, const float* __restrict__ tho,
           float* __restrict__ out) {
  __shared__ float sWh[NG][NQ];   // h-part weights (rows 512..517 of each W)
  __shared__ float sb[NG];        // biases
  __shared__ float sth[NG];       // thetas

  const int tid = threadIdx.x;    // 0..255 == batch element
  if (tid < NG) {
    int gate = tid / NQ, wire = tid % NQ;
    const float* W  = (gate == 0) ? Wf  : (gate == 1) ? Wi  : (gate == 2) ? Wu  : Wo;
    const float* bi_ = (gate == 0) ? bf  : (gate == 1) ? bi  : (gate == 2) ? bu  : bo;
    const float* th = (gate == 0) ? thf : (gate == 1) ? thi : (gate == 2) ? thu : tho;
    sb[tid]  = bi_[wire];
    sth[tid] = th[wire];
#pragma unroll
    for (int j = 0; j < NQ; ++j) sWh[tid][j] = W[(DD + j) * NQ + wire];
  }
  __syncthreads();

  float h[NQ], c[NQ];
#pragma unroll
  for (int w = 0; w < NQ; ++w) { h[w] = 0.0f; c[w] = 0.0f; }

  for (int t = 0; t < TT; ++t) {
    const float* prow = P + (size_t)(t * BB + tid) * NPAD;

    // pre[g] = x-part + bias + h @ Wh + theta  (theta folded in: a = pre+theta)
    float a[NG];
#pragma unroll
    for (int g = 0; g < NG; ++g) {
      float v = prow[g] + sb[g] + sth[g];
#pragma unroll
      for (int j = 0; j < NQ; ++j) v += h[j] * sWh[g][j];
      a[g] = v;
    }

    // closed-form qlayer per gate
    float q[4][NQ];
#pragma unroll
    for (int gate = 0; gate < 4; ++gate) {
      float z[NQ];
#pragma unroll
      for (int w = 0; w < NQ; ++w) z[w] = cosf(a[gate * NQ + w]);
      q[gate][1] = z[0] * z[1];
      q[gate][2] = q[gate][1] * z[2];
      q[gate][3] = q[gate][2] * z[3];
      q[gate][4] = q[gate][3] * z[4];
      q[gate][5] = q[gate][4] * z[5];
      q[gate][0] = ((z[1] * z[2]) * (z[3] * z[4])) * z[5];
    }

    float* orow = out + (size_t)(t * BB + tid) * NQ;
#pragma unroll
    for (int w = 0; w < NQ; ++w) {
      float fg = sigmoidf_(q[0][w]);
      float ig = sigmoidf_(q[1][w]);
      float gg = tanhf(q[2][w]);
      float og = sigmoidf_(q[3][w]);
      c[w] = fg * c[w] + ig * gg;
      h[w] = og * tanhf(c[w]);
      orow[w] = h[w];
    }
  }

  // final states: hx then cx, appended after outputs
  float* hx = out + (size_t)TT * BB * NQ;
  float* cx = hx + (size_t)BB * NQ;
#pragma unroll
  for (int w = 0; w < NQ; ++w) {
    hx[tid * NQ + w] = h[w];
    cx[tid * NQ + w] = c[w];
  }
}

// ---------------------------------------------------------------------------
extern "C" void kernel_launch(void* const* d_in, const int* in_sizes, int n_in,
                              void* d_out, int out_size, void* d_ws, size_t ws_size,
                              hipStream_t stream) {
  const float* X   = (const float*)d_in[0];   // (64,256,512)
  const float* Wf  = (const float*)d_in[1];   // (518,6)
  const float* bf  = (const float*)d_in[2];   // (6,)
  const float* Wi  = (const float*)d_in[3];
  const float* bi  = (const float*)d_in[4];
  const float* Wu  = (const float*)d_in[5];
  const float* bu  = (const float*)d_in[6];
  const float* Wo  = (const float*)d_in[7];
  const float* bo  = (const float*)d_in[8];
  const float* thf = (const float*)d_in[9];
  const float* thi = (const float*)d_in[10];
  const float* thu = (const float*)d_in[11];
  const float* tho = (const float*)d_in[12];

  float* Bt = (float*)d_ws;                   // 32*512   floats (64 KB)
  float* P  = Bt + NPAD * DD;                 // 16384*32 floats (2 MB)

  // 1) pack transposed/padded x-part weights
  qlstm_pack_w<<<(NPAD * DD) / 256, 256, 0, stream>>>(Wf, Wi, Wu, Wo, Bt);

  // 2) bulk GEMM: 1024 waves, 4 waves/block -> 256 blocks of 128 threads
  qlstm_gemm<<<MROWS / 16 / 4, 128, 0, stream>>>(X, Bt, P);

  // 3) sequential scan: one block, one thread per batch element
  qlstm_scan<<<1, BB, 0, stream>>>(P, Wf, bf, Wi, bi, Wu, bu, Wo, bo,
                                   thf, thi, thu, tho, (float*)d_out);
}